// AliasFreeActivation_1331439862241
// MI455X (gfx1250) — compile-verified
//
#include <hip/hip_runtime.h>
#include <stdint.h>

typedef float v2f __attribute__((ext_vector_type(2)));
typedef float v8f __attribute__((ext_vector_type(8)));

#define LRELU_SLOPE 0.2f
#define ACT_SCALE   1.41421356237309515f  // sqrt(2)

// LDS pitches
#define XP  68   // X: 64 x 68  (272B rows: 16B-aligned for async b128; 68%64==4 -> conflict-free)
#define T1P 33   // T1cb: 64 x 33
#define T2P 33   // T2cb: 128 x 33
#define T3P 65   // T3:   128 x 65

#define X_FLOATS  (64 * XP)    // 4352
#define T1_FLOATS (64 * T1P)   // 2112
#define T2_FLOATS (128 * T2P)  // 4224
#define T3_FLOATS (128 * T3P)  // 8320 (> T1+T2, region reuse)
#define TAP_PAD   40           // taps live at [40..52) of a 96-entry zeroed table
#define SMEM_FLOATS (X_FLOATS + T3_FLOATS + 192)

// Branchless banded-matrix elements via zero-padded tap tables.
// U[r][c] = upf2[r-2c+5], D[r][c] = dwf[2r-c+6]; index t+TAP_PAD stays in [-1,95]
// for every k actually read (fixed blocks overrun the band by a bounded amount,
// and overrun reads return the stored zeros).
__device__ __forceinline__ float uelem(const float* s_upP, int r, int c) {
  return s_upP[r - 2 * c + 5 + TAP_PAD];
}
__device__ __forceinline__ float delem(const float* s_dwP, int r, int c) {
  return s_dwP[2 * r - c + 6 + TAP_PAD];
}

// fp32 WMMA: D(16x16) = A(16x4) * B(4x16) + C -> v_wmma_f32_16x16x4_f32
__device__ __forceinline__ v8f wmma4(v2f a, v2f b, v8f c) {
  return __builtin_amdgcn_wmma_f32_16x16x4_f32(
      false, a, false, b, (short)0, c, false, false);
}

// Aligned, clamped base of a 5-block window covering the U band for rows [r0,r0+15]
__device__ __forceinline__ int ubase(int r0) {
  int kb = r0 - 6; if (kb < 0) kb = 0;
  kb = (kb >> 1) & ~3;
  if (kb > 44) kb = 44;          // reads kb..kb+19 <= 63
  return kb;
}

__global__ __launch_bounds__(256)
void afa_kernel(const float* __restrict__ in, const float* __restrict__ bias,
                const float* __restrict__ upf, const float* __restrict__ dwf,
                float* __restrict__ out, int C) {
  __shared__ float smem[SMEM_FLOATS];
  float* s_x   = smem;                         // 64 x XP
  float* s_t1  = smem + X_FLOATS;              // 64 x T1P
  float* s_t2  = s_t1 + T1_FLOATS;             // 128 x T2P
  float* s_t3  = smem + X_FLOATS;              // 128 x T3P (overlaps t1+t2; phased)
  float* s_upP = smem + X_FLOATS + T3_FLOATS;  // 96-entry padded table
  float* s_dwP = s_upP + 96;                   // 96-entry padded table

  const int tid  = threadIdx.x;
  const int lane = tid & 31;
  // Scalar wave id -> all tile coords / k-bases are SALU; EXEC stays full for WMMA.
  const int wave = __builtin_amdgcn_readfirstlane(tid >> 5);
  const int nc = blockIdx.x;
  const int c  = nc % C;

  // Build zero-padded tap tables in one race-free pass.
  if (tid < 192) {
    const int half = (tid >= 96);
    const int j = tid - (half ? 96 : 0) - TAP_PAD;
    float v = 0.0f;
    if (j >= 0 && j < 12) v = half ? dwf[j] : 2.0f * upf[j];  // UP gain folded in
    smem[X_FLOATS + T3_FLOATS + tid] = v;
  }

  // ---- async global->LDS load of the 64x64 tile (b128 per lane) ----
  const float* gin = in + (size_t)nc * 4096;
  {
    const unsigned sx_base = (unsigned)(uintptr_t)s_x;  // LDS byte offset
    const unsigned long long gbase = (unsigned long long)(uintptr_t)gin;
    #pragma unroll
    for (int j = 0; j < 4; ++j) {
      const int f0  = (tid + j * 256) * 4;          // first float of this 16B chunk
      const int row = f0 >> 6, col = f0 & 63;
      const unsigned loff = sx_base + (unsigned)((row * XP + col) * 4);
      const unsigned goff = (unsigned)(f0 * 4);
      asm volatile("global_load_async_to_lds_b128 %0, %1, %2"
                   :: "v"(loff), "v"(goff), "s"(gbase) : "memory");
    }
    asm volatile("s_wait_asynccnt 0" ::: "memory");
  }
  const float bval = bias[c];   // folded into T1 A-fragments (linear pre-filter op)
  __syncthreads();

  // fragment lane decomposition (ISA 7.12.2, 32-bit A/B/C layouts)
  const int lm  = lane & 15;          // M (A) or N (B/C) within tile
  const int lk  = (lane >> 4) << 1;   // K offset: 0 / 2
  const int lmh = (lane >> 4) << 3;   // C/D row offset: 0 / 8

  v8f acc3[4] = {};                   // persistent T3 accumulators (rows 16*wave)

  for (int CB = 0; CB < 4; ++CB) {    // 32-wide column block of the 128 domain
    const int c0 = CB * 32;

    // ---- T1cb = (X+b) @ U^T  (64x32), one tile per wave, fixed 5 K-blocks ----
    {
      const int m0  = (wave & 3) * 16;        // X row block
      const int NJ  = wave >> 2;
      const int n0g = c0 + NJ * 16;           // global T1 col = U row
      const int kb  = ubase(n0g);
      v8f acc = {};
      #pragma unroll
      for (int s = 0; s < 5; ++s) {
        const int k0 = kb + 4 * s;
        v2f a; a.x = s_x[(m0 + lm) * XP + k0 + lk] + bval;
               a.y = s_x[(m0 + lm) * XP + k0 + lk + 1] + bval;
        v2f b; b.x = uelem(s_upP, n0g + lm, k0 + lk);      // B[k][n] = U[n][k]
               b.y = uelem(s_upP, n0g + lm, k0 + lk + 1);
        acc = wmma4(a, b, acc);
      }
      const int n0l = NJ * 16;
      #pragma unroll
      for (int v = 0; v < 8; ++v)
        s_t1[(m0 + lmh + v) * T1P + n0l + lm] = acc[v];
    }
    __syncthreads();

    // ---- T2cb = U @ T1cb (128x32) + fused leaky-ReLU, 2 tiles/wave, 5 K-blocks ----
    {
      const int m0 = wave * 16;               // U row block
      const int kb = ubase(m0);
      #pragma unroll
      for (int NJ = 0; NJ < 2; ++NJ) {
        const int n0l = NJ * 16;
        v8f acc = {};
        #pragma unroll
        for (int s = 0; s < 5; ++s) {
          const int k0 = kb + 4 * s;
          v2f a; a.x = uelem(s_upP, m0 + lm, k0 + lk);     // A[m][k] = U[m][k]
                 a.y = uelem(s_upP, m0 + lm, k0 + lk + 1);
          v2f b; b.x = s_t1[(k0 + lk)     * T1P + n0l + lm];
                 b.y = s_t1[(k0 + lk + 1) * T1P + n0l + lm];
          acc = wmma4(a, b, acc);
        }
        #pragma unroll
        for (int v = 0; v < 8; ++v) {         // fused activation
          float x = acc[v];
          acc[v] = (fmaxf(x, 0.0f) + LRELU_SLOPE * fminf(x, 0.0f)) * ACT_SCALE;
        }
        #pragma unroll
        for (int v = 0; v < 8; ++v)
          s_t2[(m0 + lmh + v) * T2P + n0l + lm] = acc[v];
      }
    }
    __syncthreads();

    // ---- T3 += act(T2cb) @ D^T  (128x64 accumulated across CB windows) ----
    // Window-clipped band width is genuinely variable -> scalar runtime loop.
    {
      const int m0 = wave * 16;               // T3 row block
      #pragma unroll
      for (int NJ = 0; NJ < 4; ++NJ) {
        const int n0 = NJ * 16;               // T3 col = D row
        int lo = 2 * n0 - 5 - c0;  if (lo < 0)  lo = 0;
        int hi = 2 * n0 + 36 - c0; if (hi > 31) hi = 31;
        if (lo > hi) continue;
        v8f acc = acc3[NJ];
        for (int k0 = lo & ~3; k0 <= hi; k0 += 4) {
          v2f a; a.x = s_t2[(m0 + lm) * T2P + k0 + lk];
                 a.y = s_t2[(m0 + lm) * T2P + k0 + lk + 1];
          v2f b; b.x = delem(s_dwP, n0 + lm, c0 + k0 + lk);  // B[k][n] = D[n][kg]
                 b.y = delem(s_dwP, n0 + lm, c0 + k0 + lk + 1);
          acc = wmma4(a, b, acc);
        }
        acc3[NJ] = acc;
      }
    }
    __syncthreads();  // protect t1/t2 before next CB overwrites them
  }

  // ---- spill T3 accumulators to LDS (region reuse safe past last barrier) ----
  {
    const int m0 = wave * 16;
    #pragma unroll
    for (int NJ = 0; NJ < 4; ++NJ) {
      const int n0 = NJ * 16;
      #pragma unroll
      for (int v = 0; v < 8; ++v)
        s_t3[(m0 + lmh + v) * T3P + n0 + lm] = acc3[NJ][v];
    }
  }
  __syncthreads();

  // ---- Y = D @ T3 (64x64), 2 tiles/wave, fixed 12 K-blocks, store to HBM ----
  {
    const int m0 = (wave >> 1) * 16;          // Y row = D row
    int kb = 2 * m0 - 5; if (kb < 0) kb = 0;
    kb &= ~3; if (kb > 80) kb = 80;           // reads kb..kb+47 <= 127; covers band
    float* gout = out + (size_t)nc * 4096;
    #pragma unroll
    for (int t = 0; t < 2; ++t) {
      const int n0 = ((wave & 1) * 2 + t) * 16;
      v8f acc = {};
      #pragma unroll
      for (int s = 0; s < 12; ++s) {
        const int k0 = kb + 4 * s;
        v2f a; a.x = delem(s_dwP, m0 + lm, k0 + lk);        // A[m][k] = D[m][k]
               a.y = delem(s_dwP, m0 + lm, k0 + lk + 1);
        v2f b; b.x = s_t3[(k0 + lk)     * T3P + n0 + lm];
               b.y = s_t3[(k0 + lk + 1) * T3P + n0 + lm];
        acc = wmma4(a, b, acc);
      }
      #pragma unroll
      for (int v = 0; v < 8; ++v)
        gout[(m0 + lmh + v) * 64 + n0 + lm] = acc[v];
    }
  }
}

extern "C" void kernel_launch(void* const* d_in, const int* in_sizes, int n_in,
                              void* d_out, int out_size, void* d_ws, size_t ws_size,
                              hipStream_t stream) {
  const float* in   = (const float*)d_in[0];   // (8,512,64,64) fp32
  const float* bias = (const float*)d_in[1];   // (512,)
  const float* upf  = (const float*)d_in[2];   // 12 taps
  const float* dwf  = (const float*)d_in[3];   // 12 taps
  float* out = (float*)d_out;

  const int C  = in_sizes[1];
  const int NC = in_sizes[0] / 4096;           // N*C tiles of 64x64
  afa_kernel<<<NC, 256, 0, stream>>>(in, bias, upf, dwf, out, C);
}